// AlignmentModule_49469433315430
// MI455X (gfx1250) — compile-verified
//
#include <hip/hip_runtime.h>

typedef float v2f __attribute__((ext_vector_type(2)));
typedef float v8f __attribute__((ext_vector_type(8)));

#define NEGINF (-1e9f)

constexpr int Bn = 8, Sn = 128, Tn = 512, DT = 256, DF = 80;

__device__ __forceinline__ v2f ld2(const float* p) { return *(const v2f*)p; }

// -------------------------------------------------------------------------
// Weight transpose: W[co][ci][k] -> WT[k][ci/2][co][ci&1]  (pair-interleaved
// so a B-fragment (two consecutive K rows, fixed N) is one float2 load)
// -------------------------------------------------------------------------
__global__ void k_wtrans(const float* __restrict__ W, float* __restrict__ WT,
                         int COUT, int CIN, int KW) {
  int idx = blockIdx.x * blockDim.x + threadIdx.x;
  int total = COUT * CIN * KW;
  if (idx >= total) return;
  int k  = idx % KW;
  int ci = (idx / KW) % CIN;
  int co = idx / (KW * CIN);
  WT[((size_t)(k * (CIN >> 1) + (ci >> 1)) * COUT + co) * 2 + (ci & 1)] = W[idx];
}

// -------------------------------------------------------------------------
// Conv-as-GEMM with WMMA f32 16x16x4.
// X: (B, L, CIN) channels-contiguous. Y: (B, L, 256).
// Workgroup tile: M=64 rows x N=256. 8 waves; wave w owns a 32x64 C strip:
//   mbase = (w&1)*32 (two 16-row subtiles), nbase = (w>>1)*64 (4 N-tiles).
// Per 4-K step: 2 A-frag b64 loads + 4 B-frag b64 loads -> 8 WMMAs
// (B fragments shared by both M-subtiles: 0.75 loads per WMMA).
// A frag: lane -> row (lane&15), K pair = kb + 2*(lane>>4) + {0,1}.
// -------------------------------------------------------------------------
__global__ void k_conv_gemm(const float* __restrict__ X, const float* __restrict__ WT,
                            const float* __restrict__ bias, float* __restrict__ Y,
                            int L, int CIN, int KW, int pad, int relu) {
  const int tid  = threadIdx.x;
  const int lane = tid & 31, wave = tid >> 5;
  const int lo = lane & 15, hi = lane >> 4;
  const int b  = blockIdx.y;
  const int s0 = blockIdx.x * 64;
  const int mbase = (wave & 1) * 32;
  const int nbase = (wave >> 1) * 64;
  const int CIN2  = CIN >> 1;

  v8f acc[2][4];
  #pragma unroll
  for (int u = 0; u < 2; ++u)
    #pragma unroll
    for (int nt = 0; nt < 4; ++nt)
      #pragma unroll
      for (int r = 0; r < 8; ++r) acc[u][nt][r] = 0.f;

  const int arow = s0 + mbase + lo;
  const v2f vzero = {0.f, 0.f};

  for (int k = 0; k < KW; ++k) {
    const int  pos0 = arow + k - pad;
    const int  pos1 = pos0 + 16;
    const bool ok0  = (pos0 >= 0) && (pos0 < L);
    const bool ok1  = (pos1 >= 0) && (pos1 < L);
    const float* ap0 = X + ((size_t)(b * L + pos0) * CIN + 2 * hi);
    const float* ap1 = X + ((size_t)(b * L + pos1) * CIN + 2 * hi);
    const float* wp  = WT + (size_t)(k * CIN2 + hi) * 512;  // *256*2
    for (int kb = 0; kb < CIN; kb += 4) {
      v2f a0 = vzero, a1 = vzero;
      if (ok0) a0 = ld2(ap0 + kb);
      if (ok1) a1 = ld2(ap1 + kb);
      const float* wrow = wp + (size_t)(kb >> 1) * 512;
      #pragma unroll
      for (int nt = 0; nt < 4; ++nt) {
        v2f w2 = ld2(wrow + (nbase + nt * 16 + lo) * 2);
        acc[0][nt] = __builtin_amdgcn_wmma_f32_16x16x4_f32(
            false, a0, false, w2, (short)0, acc[0][nt], false, false);
        acc[1][nt] = __builtin_amdgcn_wmma_f32_16x16x4_f32(
            false, a1, false, w2, (short)0, acc[1][nt], false, false);
      }
    }
  }

  #pragma unroll
  for (int nt = 0; nt < 4; ++nt) {
    int col = nbase + nt * 16 + lo;
    float bv = bias[col];
    #pragma unroll
    for (int u = 0; u < 2; ++u) {
      #pragma unroll
      for (int r = 0; r < 8; ++r) {
        int row = s0 + mbase + u * 16 + r + 8 * hi;  // C layout: VGPR r = M r / r+8
        float v = acc[u][nt][r] + bv;
        if (relu) v = fmaxf(v, 0.f);
        Y[(size_t)(b * L + row) * 256 + col] = v;
      }
    }
  }
}

// -------------------------------------------------------------------------
// Row L2-norm^2 over 256 channels. One wave per row.
// -------------------------------------------------------------------------
__global__ void k_rownorm(const float* __restrict__ X, float* __restrict__ N2, int nrows) {
  int row  = blockIdx.x * 4 + (threadIdx.x >> 5);
  int lane = threadIdx.x & 31;
  if (row >= nrows) return;
  const float* p = X + (size_t)row * 256;
  float s = 0.f;
  for (int c = lane; c < 256; c += 32) { float v = p[c]; s += v * v; }
  for (int off = 16; off > 0; off >>= 1) s += __shfl_xor(s, off, 32);
  if (lane == 0) N2[row] = s;
}

__global__ void k_zero(float* __restrict__ p, int n) {
  int i = blockIdx.x * blockDim.x + threadIdx.x;
  if (i < n) p[i] = 0.f;
}

// -------------------------------------------------------------------------
// dist + log_softmax over S (axis 1), fused with the cross-term GEMM.
// Workgroup = 8 waves; per batch b and 64-col T tile. Wave w owns S rows
// [16w,16w+16) x 64 cols. dot via WMMA; dist = sqrt(max(h2+m2-2dot,0));
// masked rows (s >= tok_len) -> dist 0 (matches reference). Column softmax
// reduced across waves through staged LDS (deterministic, no atomics).
// Writes log_a_soft (B,S,T) to d_out and a transposed copy (B,T,S) to ws.
// -------------------------------------------------------------------------
__global__ void k_dist_softmax(const float* __restrict__ HT, const float* __restrict__ MT,
                               const float* __restrict__ H2, const float* __restrict__ M2,
                               const int* __restrict__ tok_len,
                               float* __restrict__ lsoft, float* __restrict__ lpT) {
  __shared__ float stage[8][2][64];
  __shared__ float cred[64];

  const int tid  = threadIdx.x;
  const int lane = tid & 31, wave = tid >> 5;
  const int lo = lane & 15, hi = lane >> 4;
  const int b = blockIdx.y, t0 = blockIdx.x * 64;
  const int srow = wave * 16;

  v8f acc[4];
  #pragma unroll
  for (int nt = 0; nt < 4; ++nt)
    #pragma unroll
    for (int r = 0; r < 8; ++r) acc[nt][r] = 0.f;

  const float* ap = HT + ((size_t)(b * Sn + srow + lo) * DT + 2 * hi);
  const float* bp0 = MT + ((size_t)(b * Tn + t0 + lo) * DT + 2 * hi);
  for (int kb = 0; kb < DT; kb += 4) {
    v2f a = ld2(ap + kb);
    #pragma unroll
    for (int nt = 0; nt < 4; ++nt) {
      v2f bb = ld2(bp0 + (size_t)nt * 16 * DT + kb);
      acc[nt] = __builtin_amdgcn_wmma_f32_16x16x4_f32(
          false, a, false, bb, (short)0, acc[nt], false, false);
    }
  }

  const int tok = tok_len[b];
  float h2[8];
  #pragma unroll
  for (int r = 0; r < 8; ++r) h2[r] = H2[b * Sn + srow + r + 8 * hi];

  #pragma unroll
  for (int nt = 0; nt < 4; ++nt) {
    int col = t0 + nt * 16 + lo;
    float m2 = M2[b * Tn + col];
    #pragma unroll
    for (int r = 0; r < 8; ++r) {
      int row = srow + r + 8 * hi;
      float dd   = h2[r] + m2 - 2.f * acc[nt][r];
      float dist = sqrtf(fmaxf(dd, 0.f));
      acc[nt][r] = (row < tok) ? -dist : 0.f;
    }
  }

  // column max (staged per wave/half, then tree-reduced: deterministic)
  #pragma unroll
  for (int nt = 0; nt < 4; ++nt) {
    int c = nt * 16 + lo;
    float pm = acc[nt][0];
    #pragma unroll
    for (int r = 1; r < 8; ++r) pm = fmaxf(pm, acc[nt][r]);
    stage[wave][hi][c] = pm;
  }
  __syncthreads();
  if (tid < 64) {
    float mx = -3.4e38f;
    for (int w = 0; w < 8; ++w)
      for (int h = 0; h < 2; ++h) mx = fmaxf(mx, stage[w][h][tid]);
    cred[tid] = mx;
  }
  __syncthreads();
  // column sum of exp
  #pragma unroll
  for (int nt = 0; nt < 4; ++nt) {
    int c = nt * 16 + lo;
    float mx = cred[c];
    float es = 0.f;
    #pragma unroll
    for (int r = 0; r < 8; ++r) es += expf(acc[nt][r] - mx);
    stage[wave][hi][c] = es;
  }
  __syncthreads();
  if (tid < 64) {
    float s = 0.f;
    for (int w = 0; w < 8; ++w)
      for (int h = 0; h < 2; ++h) s += stage[w][h][tid];
    cred[tid] = cred[tid] + logf(s);   // lse per column
  }
  __syncthreads();
  #pragma unroll
  for (int nt = 0; nt < 4; ++nt) {
    int c = nt * 16 + lo;
    int col = t0 + c;
    float lse = cred[c];
    #pragma unroll
    for (int r = 0; r < 8; ++r) {
      int row = srow + r + 8 * hi;
      float v = acc[nt][r] - lse;
      lsoft[(size_t)(b * Sn + row) * Tn + col] = v;
      lpT  [(size_t)(b * Tn + col) * Sn + row] = v;
    }
  }
}

// -------------------------------------------------------------------------
// MAS: one wave per batch. Lane l holds s = 4l..4l+3 (q0..q3).
// Forward: 511 steps, one shfl_up per step; choice bits packed via ballot:
// bit l of ch[t][k] <-> s = 4l + k. Backtrack on lane 0; duration counts
// accumulated in LDS and written as d.
// -------------------------------------------------------------------------
__global__ void k_mas(const float* __restrict__ lpT, const int* __restrict__ tok_len,
                      const int* __restrict__ feat_len,
                      float* __restrict__ dd, float* __restrict__ ahard) {
  __shared__ unsigned ch[Tn][4];
  __shared__ int cnt[Sn];
  const int l = threadIdx.x;
  const int b = blockIdx.x;

  for (int i = l; i < Sn; i += 32) cnt[i] = 0;
  if (l < 4) ch[0][l] = 0u;

  const float4* LP = (const float4*)(lpT + (size_t)b * Tn * Sn);
  float q0, q1, q2, q3;
  {
    float4 lp = LP[l];
    q0 = (l == 0) ? lp.x : NEGINF;
    q1 = NEGINF; q2 = NEGINF; q3 = NEGINF;
  }
  for (int t = 1; t < Tn; ++t) {
    float4 lp = LP[t * 32 + l];
    float prev = __shfl_up(q3, 1, 32);
    if (l == 0) prev = NEGINF;
    bool t0b = prev >= q0, t1b = q0 >= q1, t2b = q1 >= q2, t3b = q2 >= q3;
    float n0 = lp.x + fmaxf(q0, prev);
    float n1 = lp.y + fmaxf(q1, q0);
    float n2 = lp.z + fmaxf(q2, q1);
    float n3 = lp.w + fmaxf(q3, q2);
    unsigned b0 = (unsigned)__ballot(t0b);
    unsigned b1 = (unsigned)__ballot(t1b);
    unsigned b2 = (unsigned)__ballot(t2b);
    unsigned b3 = (unsigned)__ballot(t3b);
    if (l == 0) { ch[t][0] = b0; ch[t][1] = b1; ch[t][2] = b2; ch[t][3] = b3; }
    q0 = n0; q1 = n1; q2 = n2; q3 = n3;
  }
  __syncthreads();

  if (l == 0) {
    int i  = tok_len[b] - 1;
    int fl = feat_len[b];
    for (int t = Tn - 1; t >= 0; --t) {
      if (t < fl) {
        ahard[(size_t)(b * Sn + i) * Tn + t] = 1.0f;
        cnt[i] += 1;
        if (t > 0) {
          unsigned bits = ch[t][i & 3];
          i -= (int)((bits >> (i >> 2)) & 1u);
        }
      }
    }
  }
  __syncthreads();
  for (int s = l; s < Sn; s += 32) dd[b * Sn + s] = (float)cnt[s];
}

// -------------------------------------------------------------------------
extern "C" void kernel_launch(void* const* d_in, const int* in_sizes, int n_in,
                              void* d_out, int out_size, void* d_ws, size_t ws_size,
                              hipStream_t stream) {
  const float* h    = (const float*)d_in[0];
  const float* m    = (const float*)d_in[1];
  const int*   tokl = (const int*)d_in[2];
  const int*   fetl = (const int*)d_in[3];
  // d_in[4] = mask (unused; recomputed from token_length)
  const float* W1 = (const float*)d_in[5];
  const float* b1 = (const float*)d_in[6];
  const float* W2 = (const float*)d_in[7];
  const float* b2 = (const float*)d_in[8];
  const float* F1 = (const float*)d_in[9];
  const float* g1 = (const float*)d_in[10];
  const float* F2 = (const float*)d_in[11];
  const float* g2 = (const float*)d_in[12];
  const float* F3 = (const float*)d_in[13];
  const float* g3 = (const float*)d_in[14];

  float* ws = (float*)d_ws;
  size_t off = 0;
  float* w1t = ws + off; off += (size_t)3 * 128 * 256 * 2;
  float* w2t = ws + off; off += (size_t)1 * 128 * 256 * 2;
  float* f1t = ws + off; off += (size_t)3 * 40  * 256 * 2;
  float* f2t = ws + off; off += (size_t)3 * 128 * 256 * 2;
  float* f3t = ws + off; off += (size_t)1 * 128 * 256 * 2;
  float* t1  = ws + off; off += (size_t)Bn * Sn * 256;
  float* ht  = ws + off; off += (size_t)Bn * Sn * 256;
  float* z1  = ws + off; off += (size_t)Bn * Tn * 256;
  float* z2  = ws + off; off += (size_t)Bn * Tn * 256;
  float* mt  = z1;  // z1 dead after F2 -> reuse for F3 output
  float* h2  = ws + off; off += (size_t)Bn * Sn;
  float* m2  = ws + off; off += (size_t)Bn * Tn;
  float* lpT = ws + off; off += (size_t)Bn * Tn * Sn;

  float* out  = (float*)d_out;
  float* d_d  = out;                                   // (B,S)
  float* d_ls = out + Bn * Sn;                         // (B,S,T)
  float* d_ah = out + Bn * Sn + (size_t)Bn * Sn * Tn;  // (B,S,T)

  auto tr = [&](const float* W, float* WT, int COUT, int CIN, int KW) {
    int total = COUT * CIN * KW;
    hipLaunchKernelGGL(k_wtrans, dim3((total + 255) / 256), dim3(256), 0, stream,
                       W, WT, COUT, CIN, KW);
  };
  tr(W1, w1t, 256, 256, 3);
  tr(W2, w2t, 256, 256, 1);
  tr(F1, f1t, 256, 80, 3);
  tr(F2, f2t, 256, 256, 3);
  tr(F3, f3t, 256, 256, 1);

  // text: conv3+relu -> conv1   (M tiles of 64 rows)
  hipLaunchKernelGGL(k_conv_gemm, dim3(Sn / 64, Bn), dim3(256), 0, stream,
                     h, w1t, b1, t1, Sn, 256, 3, 1, 1);
  hipLaunchKernelGGL(k_conv_gemm, dim3(Sn / 64, Bn), dim3(256), 0, stream,
                     t1, w2t, b2, ht, Sn, 256, 1, 0, 0);
  // feat: conv3 (relu at store, consumed relu'd) -> conv3+relu -> conv1
  hipLaunchKernelGGL(k_conv_gemm, dim3(Tn / 64, Bn), dim3(256), 0, stream,
                     m, f1t, g1, z1, Tn, 80, 3, 1, 1);
  hipLaunchKernelGGL(k_conv_gemm, dim3(Tn / 64, Bn), dim3(256), 0, stream,
                     z1, f2t, g2, z2, Tn, 256, 3, 1, 1);
  hipLaunchKernelGGL(k_conv_gemm, dim3(Tn / 64, Bn), dim3(256), 0, stream,
                     z2, f3t, g3, mt, Tn, 256, 1, 0, 0);

  hipLaunchKernelGGL(k_rownorm, dim3(Bn * Sn / 4), dim3(128), 0, stream, ht, h2, Bn * Sn);
  hipLaunchKernelGGL(k_rownorm, dim3(Bn * Tn / 4), dim3(128), 0, stream, mt, m2, Bn * Tn);

  int nA = Bn * Sn * Tn;
  hipLaunchKernelGGL(k_zero, dim3((nA + 255) / 256), dim3(256), 0, stream, d_ah, nA);

  hipLaunchKernelGGL(k_dist_softmax, dim3(Tn / 64, Bn), dim3(256), 0, stream,
                     ht, mt, h2, m2, tokl, d_ls, lpT);

  hipLaunchKernelGGL(k_mas, dim3(Bn), dim3(32), 0, stream, lpT, tokl, fetl, d_d, d_ah);
}